// PointsRasterizer_23081154249382
// MI455X (gfx1250) — compile-verified
//
#include <hip/hip_runtime.h>

namespace {
constexpr int   kS   = 128;           // image size
constexpr int   kP   = 2048;          // points per batch
constexpr int   kN   = 2;             // batches
constexpr int   kK   = 8;             // points per pixel
constexpr float kR2  = 0.01f * 0.01f; // radius^2
constexpr float kPre = 4e-6f;         // prefilter slack (expanded-form rounding)
constexpr float kBig = 1e10f;
}

typedef __attribute__((ext_vector_type(2))) float v2f;
typedef __attribute__((ext_vector_type(8))) float v8f;

__global__ __launch_bounds__(256) void
PointsRasterizer_23081154249382_kernel(const float* __restrict__ pts,
                                       float* __restrict__ out)
{
    // SoA LDS staging (56 KB total):
    //   sfrag[2q+0] = (1, y)        A-fragment for lanes 0-15  (K = 0,1)
    //   sfrag[2q+1] = (x, x^2+y^2)  A-fragment for lanes 16-31 (K = 2,3)
    //   sxy[q]      = (x, y)        exact-recompute path
    //   sz[q]       = z
    __shared__ float2 sfrag[2 * kP];
    __shared__ float2 sxy[kP];
    __shared__ float  sz[kP];

    const int tid = threadIdx.x;
    const int n   = blockIdx.x / kS;   // batch
    const int row = blockIdx.x % kS;   // image row (y / H axis)
    const float* bp = pts + (size_t)n * kP * 3;

    for (int q = tid; q < kP; q += 256) {
        float x = bp[q * 3 + 0];
        float y = bp[q * 3 + 1];
        float z = bp[q * 3 + 2];
        sfrag[2 * q + 0] = make_float2(1.0f, y);
        sfrag[2 * q + 1] = make_float2(x, x * x + y * y);
        sxy[q] = make_float2(x, y);
        sz[q]  = z;
    }
    __syncthreads();

    const int wave = tid >> 5;
    const int lane = tid & 31;
    const int half = lane >> 4;        // 0: points 0-7 of chunk, 1: points 8-15
    const int pixn = lane & 15;
    const int col  = wave * 16 + pixn; // image column (x / W axis)

    const float gy = 1.0f - 2.0f * ((float)row + 0.5f) * (1.0f / (float)kS);
    const float gx = 1.0f - 2.0f * ((float)col + 0.5f) * (1.0f / (float)kS);

    // B fragment = per-pixel features (loop invariant).
    // K=0: gy^2+gx^2   K=1: -2gy   K=2: -2gx   K=3: 1
    // f32 B 4x16 layout: lanes 0-15 hold K={0,1}; lanes 16-31 hold K={2,3}.
    v2f bfrag;
    bfrag.x = half ? (-2.0f * gx) : (gy * gy + gx * gx);
    bfrag.y = half ? 1.0f         : (-2.0f * gy);

    // Private top-8 (z ascending, point index), covering this lane's point half.
    float zk[kK];
    int   pk[kK];
    #pragma unroll
    for (int t = 0; t < kK; ++t) { zk[t] = kBig; pk[t] = -1; }

    for (int pc = 0; pc < kP; pc += 16) {
        float2 f = sfrag[2 * (pc + pixn) + half];   // one aligned ds_load_b64
        v2f afrag;
        afrag.x = f.x;
        afrag.y = f.y;

        v8f c = {};
        // D[M=point, N=pixel]: lane holds pixel (lane&15), points pc + half*8 + v.
        c = __builtin_amdgcn_wmma_f32_16x16x4_f32(false, afrag, false, bfrag,
                                                  (short)0, c, false, false);

        // Single branch per chunk: any of the 8 d2 values below prefilter?
        float m = fminf(fminf(fminf(c[0], c[1]), fminf(c[2], c[3])),
                        fminf(fminf(c[4], c[5]), fminf(c[6], c[7])));
        if (m < kR2 + kPre) {                       // rare path
            #pragma unroll
            for (int v = 0; v < 8; ++v) {
                if (c[v] < kR2 + kPre) {
                    int p = pc + half * 8 + v;
                    float2 qxy = sxy[p];
                    float  zq  = sz[p];
                    float ddy = gy - qxy.y;
                    float ddx = gx - qxy.x;
                    float d2  = ddy * ddy + ddx * ddx;  // exact reference form
                    if (d2 < kR2 && zq > 0.0f) {
                        float zz = zq; int pp = p;
                        #pragma unroll
                        for (int t = 0; t < kK; ++t) {  // branchless cmov sort
                            bool  cl  = zz < zk[t];
                            float lo  = cl ? zz    : zk[t];
                            float hi  = cl ? zk[t] : zz;
                            int   plo = cl ? pp    : pk[t];
                            int   phi = cl ? pk[t] : pp;
                            zk[t] = lo; pk[t] = plo;
                            zz = hi;    pp = phi;
                        }
                    }
                }
            }
        }
    }

    // Merge the two half-lists of each pixel (lanes L and L+16):
    // snapshot partner registers first, then branchless stable-insert
    // (tie-break on smaller point index, matching top_k stability).
    float oz[kK]; int op[kK];
    #pragma unroll
    for (int t = 0; t < kK; ++t) {
        oz[t] = __shfl_xor(zk[t], 16, 32);
        op[t] = __shfl_xor(pk[t], 16, 32);
    }
    #pragma unroll
    for (int t = 0; t < kK; ++t) {
        float zz = oz[t]; int pp = op[t];
        #pragma unroll
        for (int u = 0; u < kK; ++u) {
            bool  cl  = (zz < zk[u]) ||
                        ((zz == zk[u]) && (zz < kBig) && (pp < pk[u]));
            float lo  = cl ? zz    : zk[u];
            float hi  = cl ? zk[u] : zz;
            int   plo = cl ? pp    : pk[u];
            int   phi = cl ? pk[u] : pp;
            zk[u] = lo; pk[u] = plo;
            zz = hi;    pp = phi;
        }
    }

    if (lane < 16) {
        const size_t nhwk = (size_t)kN * kS * kS * kK;
        size_t base = (((size_t)n * kS + row) * kS + col) * kK;

        float idxv[kK], zbv[kK], dsv[kK];
        #pragma unroll
        for (int t = 0; t < kK; ++t) {
            if (pk[t] >= 0) {
                float2 qxy = sxy[pk[t]];
                float ddy = gy - qxy.y;
                float ddx = gx - qxy.x;
                idxv[t] = (float)pk[t];
                zbv[t]  = zk[t];
                dsv[t]  = ddy * ddy + ddx * ddx;   // bit-exact reference form
            } else {
                idxv[t] = -1.0f; zbv[t] = -1.0f; dsv[t] = -1.0f;
            }
        }

        float4* oi = (float4*)(out + base);
        float4* ob = (float4*)(out + nhwk + base);
        float4* od = (float4*)(out + 2 * nhwk + base);
        oi[0] = make_float4(idxv[0], idxv[1], idxv[2], idxv[3]);
        oi[1] = make_float4(idxv[4], idxv[5], idxv[6], idxv[7]);
        ob[0] = make_float4(zbv[0], zbv[1], zbv[2], zbv[3]);
        ob[1] = make_float4(zbv[4], zbv[5], zbv[6], zbv[7]);
        od[0] = make_float4(dsv[0], dsv[1], dsv[2], dsv[3]);
        od[1] = make_float4(dsv[4], dsv[5], dsv[6], dsv[7]);
    }
}

extern "C" void kernel_launch(void* const* d_in, const int* in_sizes, int n_in,
                              void* d_out, int out_size, void* d_ws, size_t ws_size,
                              hipStream_t stream)
{
    (void)in_sizes; (void)n_in; (void)out_size; (void)d_ws; (void)ws_size;
    const float* pts = (const float*)d_in[0];
    float*       out = (float*)d_out;
    // One block per image row per batch: 256 blocks x 256 threads (8 waves).
    PointsRasterizer_23081154249382_kernel<<<dim3(kN * kS), dim3(256), 0, stream>>>(pts, out);
}